// _AttentionAggBase_40321152974892
// MI455X (gfx1250) — compile-verified
//
#include <hip/hip_runtime.h>
#include <hip/hip_bf16.h>
#include <math.h>

typedef __attribute__((ext_vector_type(2))) float v2f;
typedef __attribute__((ext_vector_type(8))) float v8f;

#define DIMS 64

// ---------------------------------------------------------------------------
// Workspace layout (floats): scoreEx[E] | smax[N] | denom[N] | Mv[N*64]
// N is only known on device (dim_size input) -> every kernel derives pointers
// from (ws, E, N) itself; host only needs E for grid sizing.
// ---------------------------------------------------------------------------

__device__ __forceinline__ void atomic_max_float(float* addr, float val) {
  // ordered-int trick; works with init = 0xFF800000 (-inf)
  int iv = __float_as_int(val);
  if (iv >= 0) atomicMax((int*)addr, iv);
  else         atomicMin((unsigned int*)addr, (unsigned int)iv);
}

__global__ void k0_init(float* ws, int E, const int* __restrict__ dim_size) {
  int N = dim_size[0];
  float* base = ws + E;                 // smax | denom | Mv contiguous
  long total = (long)N * (2 + DIMS);
  long stride = (long)gridDim.x * blockDim.x;
  for (long t = (long)blockIdx.x * blockDim.x + threadIdx.x; t < total; t += stride) {
    base[t] = (t < N) ? __int_as_float((int)0xFF800000u) : 0.0f;
  }
}

// Pass 1: score[e] = M[e,:] . a  via V_WMMA_F32_16X16X4_F32 (16 edges / wave),
// then segment-max into smax[dest[e]].
__global__ void k1_score(const float* __restrict__ M, const float* __restrict__ a,
                         const int* __restrict__ dest, float* ws, int E) {
  float* scoreEx = ws;
  float* smax    = ws + E;
  int lane = threadIdx.x & 31;
  int wave = threadIdx.x >> 5;
  int tile = blockIdx.x * 8 + wave;     // 8 waves / block
  int e0 = tile * 16;
  if (e0 >= E) return;

  if (e0 + 16 <= E) {
    // Full tile: wave-uniform branch, EXEC all-1s for WMMA.
    // A (16x4 f32): lane m<16 holds row m, K=kk+{0,1}; lane m+16 holds row m, K=kk+{2,3}.
    // B (4x16 f32): broadcast a[] into every column -> same K indexing per lane.
    int row  = lane & 15;
    int koff = (lane < 16) ? 0 : 2;
    const float* Mrow = M + (size_t)(e0 + row) * DIMS;
    v8f acc = {};
#pragma unroll
    for (int kk = 0; kk < DIMS; kk += 4) {
      v2f Aop, Bop;
      Aop.x = Mrow[kk + koff];
      Aop.y = Mrow[kk + koff + 1];
      Bop.x = a[kk + koff];
      Bop.y = a[kk + koff + 1];
      acc = __builtin_amdgcn_wmma_f32_16x16x4_f32(
          /*neg_a=*/false, Aop, /*neg_b=*/false, Bop,
          /*c_mod=*/(short)0, acc, /*reuse_a=*/false, /*reuse_b=*/false);
    }
    // D layout: lane 0 holds rows 0..7 in acc[0..7]; lane 16 holds rows 8..15.
    if ((lane & 15) == 0) {
      int base = (lane >> 4) * 8;
#pragma unroll
      for (int r = 0; r < 8; ++r) {
        float s = acc[r];
        int e = e0 + base + r;
        scoreEx[e] = s;
        atomic_max_float(&smax[dest[e]], s);
      }
    }
  } else {
    // Partial tail tile: scalar dot product, one edge per lane.
    int e = e0 + lane;
    if (lane < 16 && e < E) {
      const float* Mr = M + (size_t)e * DIMS;
      float s = 0.f;
#pragma unroll
      for (int k = 0; k < DIMS; ++k) s += Mr[k] * a[k];
      scoreEx[e] = s;
      atomic_max_float(&smax[dest[e]], s);
    }
  }
}

// Pass 2: ex = exp(score - smax[dest]); denom[dest] += ex
__global__ void k2_exp(const int* __restrict__ dest, float* ws, int E,
                       const int* __restrict__ dim_size) {
  int N = dim_size[0];
  int e = blockIdx.x * blockDim.x + threadIdx.x;
  if (e >= E) return;
  float* scoreEx = ws;
  float* smax    = ws + E;
  float* denom   = ws + E + N;
  int dn = dest[e];
  float ex = expf(scoreEx[e] - smax[dn]);
  scoreEx[e] = ex;
  atomicAdd(&denom[dn], ex);
}

// Pass 3a: alpha = ex / denom[dest] (in place)
__global__ void k3_alpha(const int* __restrict__ dest, float* ws, int E,
                         const int* __restrict__ dim_size) {
  int N = dim_size[0];
  int e = blockIdx.x * blockDim.x + threadIdx.x;
  if (e >= E) return;
  float* scoreEx = ws;
  float* denom   = ws + E + N;
  scoreEx[e] = scoreEx[e] / denom[dest[e]];
}

// Pass 3b: Mv[dest[e], k] += alpha[e] * M[e, k]  (Mv is L2-resident: 12.8 MB)
__global__ void k3_scatter(const float* __restrict__ M, const int* __restrict__ dest,
                           float* ws, int E, const int* __restrict__ dim_size) {
  int N = dim_size[0];
  int t = blockIdx.x * blockDim.x + threadIdx.x;   // E*64 = 51.2M < 2^31
  if (t >= E * DIMS) return;
  int e = t >> 6;
  int k = t & (DIMS - 1);
  const float* alpha = ws;
  float* Mv = ws + E + 2 * (long)N;
  atomicAdd(&Mv[(size_t)dest[e] * DIMS + k], alpha[e] * M[t]);
}

// Pass 4: out[e, k] = Mv[src[e], k] - alpha[rev[e]] * M[rev[e], k]
__global__ void k4_out(const float* __restrict__ M, const int* __restrict__ srcIdx,
                       const int* __restrict__ revIdx, float* __restrict__ out,
                       const float* ws, int E, const int* __restrict__ dim_size) {
  int N = dim_size[0];
  int t = blockIdx.x * blockDim.x + threadIdx.x;
  if (t >= E * DIMS) return;
  int e = t >> 6;
  int k = t & (DIMS - 1);
  const float* alpha = ws;
  const float* Mv = ws + E + 2 * (long)N;
  int s = srcIdx[e];
  int r = revIdx[e];
  out[t] = Mv[(size_t)s * DIMS + k] - alpha[r] * M[(size_t)r * DIMS + k];
}

extern "C" void kernel_launch(void* const* d_in, const int* in_sizes, int n_in,
                              void* d_out, int out_size, void* d_ws, size_t ws_size,
                              hipStream_t stream) {
  const float* M          = (const float*)d_in[0];   // [E, 64] f32
  const float* a          = (const float*)d_in[1];   // [64]    f32
  const int*   edge_index = (const int*)d_in[2];     // [2, E]
  const int*   rev_index  = (const int*)d_in[3];     // [E]
  const int*   dim_size   = (const int*)d_in[4];     // scalar N (device)
  float* out = (float*)d_out;
  float* ws  = (float*)d_ws;

  int E = in_sizes[3];                 // rev_index element count
  const int* srcIdx  = edge_index;     // row 0
  const int* destIdx = edge_index + E; // row 1

  k0_init<<<2048, 256, 0, stream>>>(ws, E, dim_size);

  int tiles = (E + 15) / 16;
  k1_score<<<(tiles + 7) / 8, 256, 0, stream>>>(M, a, destIdx, ws, E);

  int gE = (E + 255) / 256;
  k2_exp  <<<gE, 256, 0, stream>>>(destIdx, ws, E, dim_size);
  k3_alpha<<<gE, 256, 0, stream>>>(destIdx, ws, E, dim_size);

  long n = (long)E * DIMS;
  int gEd = (int)((n + 255) / 256);
  k3_scatter<<<gEd, 256, 0, stream>>>(M, destIdx, ws, E, dim_size);
  k4_out    <<<gEd, 256, 0, stream>>>(M, srcIdx, rev_index, out, ws, E, dim_size);
}